// Subject_Layer_6725918786146
// MI455X (gfx1250) — compile-verified
//
#include <hip/hip_runtime.h>

// ---------------------------------------------------------------------------
// out[b,d,t] = sum_c x[b,c,t] * weights[subjects[b],c,d]
// Shapes: x (256,273,360) f32, subjects (256) int, weights (128,273,273) f32
// Per-sample GEMM via V_WMMA_F32_16X16X4_F32 (wave32).
// Addressing: wave-uniform SGPR base + 32-bit per-lane offsets (GVS form) so
// loads lower to global_load_b32 v, v, s[..] with v_add_nc_u32 increments.
// ---------------------------------------------------------------------------

typedef __attribute__((ext_vector_type(2))) float v2f;
typedef __attribute__((ext_vector_type(8))) float v8f;

// Global (addrspace(1)) pointers -> global_load/global_store (LOADcnt only).
typedef const float __attribute__((address_space(1)))* cgfp;
typedef float       __attribute__((address_space(1)))* gfp;

constexpr int C_IN  = 273;
constexpr int C_OUT = 273;
constexpr int T     = 360;
constexpr int BATCH = 256;

constexpr int MW    = 3;            // 16x16 m-tiles per wave
constexpr int NW    = 4;            // 16x16 n-tiles per wave
constexpr int MBLK  = MW * 16;      // 48 rows (d) per wave
constexpr int NBLK  = NW * 16;      // 64 cols (t) per wave
constexpr int MGRID = (C_OUT + MBLK - 1) / MBLK;  // 6
constexpr int NGRID = (T     + NBLK - 1) / NBLK;  // 6

__global__ __launch_bounds__(32)
void subject_layer_wmma(const float* __restrict__ xg,
                        const int*   __restrict__ subjects,
                        const float* __restrict__ wg,
                        float*       __restrict__ outg)
{
    const int bid = blockIdx.x;
    const int ng  = bid % NGRID;
    const int mg  = (bid / NGRID) % MGRID;
    const int b   = bid / (NGRID * MGRID);

    const int lane = threadIdx.x & 31;
    const int half = lane >> 4;     // 0: K regs {kb,kb+1}; 1: {kb+2,kb+3}
    const int l16  = lane & 15;

    const int subj = subjects[b];   // wave-uniform -> scalar load
    cgfp wb = (cgfp)wg + (size_t)subj * C_IN * C_OUT;   // uniform SGPR base
    cgfp xb = (cgfp)xg + (size_t)b    * C_IN * T;       // uniform SGPR base

    // A fragment offsets: element (k = 2*half + r, m) at k*C_OUT + m.
    // Clamp m: rows beyond C_OUT compute garbage that is never stored
    // (output rows are independent in D = A*B + C), keeping EXEC all-ones.
    unsigned aoff[MW];
#pragma unroll
    for (int i = 0; i < MW; ++i) {
        int m = mg * MBLK + i * 16 + l16;
        if (m > C_OUT - 1) m = C_OUT - 1;
        aoff[i] = (unsigned)(2 * half) * C_OUT + (unsigned)m;
    }
    // B fragment offsets: element (k = 2*half + r, n) at k*T + n. Clamp n.
    unsigned boff[NW];
#pragma unroll
    for (int j = 0; j < NW; ++j) {
        int n = ng * NBLK + j * 16 + l16;
        if (n > T - 1) n = T - 1;
        boff[j] = (unsigned)(2 * half) * T + (unsigned)n;
    }

    v8f acc[MW][NW];
#pragma unroll
    for (int i = 0; i < MW; ++i)
#pragma unroll
        for (int j = 0; j < NW; ++j)
            acc[i][j] = v8f{};

    // ---- main K loop: k in [0, 272), all 4 k's in range ----
    constexpr int KMAIN = (C_IN / 4) * 4;   // 272
    for (int kb = 0; kb < KMAIN; kb += 4) {
        v2f av[MW];
#pragma unroll
        for (int i = 0; i < MW; ++i) {
            av[i] = v2f{ wb[aoff[i]], wb[aoff[i] + C_OUT] };
            aoff[i] += 4u * C_OUT;
        }
        v2f bv[NW];
#pragma unroll
        for (int j = 0; j < NW; ++j) {
            bv[j] = v2f{ xb[boff[j]], xb[boff[j] + T] };
            boff[j] += 4u * T;
        }
#pragma unroll
        for (int i = 0; i < MW; ++i)
#pragma unroll
            for (int j = 0; j < NW; ++j)
                acc[i][j] = __builtin_amdgcn_wmma_f32_16x16x4_f32(
                    /*neg_a=*/false, av[i],
                    /*neg_b=*/false, bv[j],
                    /*c_mod=*/(short)0, acc[i][j],
                    /*reuse_a=*/false, /*reuse_b=*/false);
    }

    // ---- K tail: only k = 272 valid (half==0, reg 0); mask OOB loads ----
    {
        v2f av[MW];
#pragma unroll
        for (int i = 0; i < MW; ++i)
            av[i] = v2f{ (half == 0) ? wb[aoff[i]] : 0.0f, 0.0f };
        v2f bv[NW];
#pragma unroll
        for (int j = 0; j < NW; ++j)
            bv[j] = v2f{ (half == 0) ? xb[boff[j]] : 0.0f, 0.0f };
#pragma unroll
        for (int i = 0; i < MW; ++i)
#pragma unroll
            for (int j = 0; j < NW; ++j)
                acc[i][j] = __builtin_amdgcn_wmma_f32_16x16x4_f32(
                    false, av[i], false, bv[j], (short)0, acc[i][j],
                    false, false);
    }

    // ---- store: VGPR r holds row m = tileBase + half*8 + r, col n = l16 ----
    gfp ob = (gfp)outg + (size_t)b * C_OUT * T;          // uniform SGPR base
#pragma unroll
    for (int i = 0; i < MW; ++i) {
        const int m0 = mg * MBLK + i * 16 + half * 8;
#pragma unroll
        for (int j = 0; j < NW; ++j) {
            const int n = ng * NBLK + j * 16 + l16;
            if (n < T) {
#pragma unroll
                for (int r = 0; r < 8; ++r) {
                    const int m = m0 + r;
                    if (m < C_OUT)
                        ob[(unsigned)m * T + (unsigned)n] = acc[i][j][r];
                }
            }
        }
    }
}

extern "C" void kernel_launch(void* const* d_in, const int* in_sizes, int n_in,
                              void* d_out, int out_size, void* d_ws, size_t ws_size,
                              hipStream_t stream)
{
    const float* x        = (const float*)d_in[0];
    const int*   subjects = (const int*)  d_in[1];
    const float* w        = (const float*)d_in[2];
    float*       out      = (float*)d_out;

    const int nblocks = BATCH * MGRID * NGRID;   // 256*6*6 = 9216 one-wave blocks
    subject_layer_wmma<<<nblocks, 32, 0, stream>>>(x, subjects, w, out);
}